// DGL_ResGatedGraphConv_12120397710134
// MI455X (gfx1250) — compile-verified
//
#include <hip/hip_runtime.h>
#include <math.h>

#define DD 128
#define NNODES 50000
#define NEDGES 500000
#define EPS_BN 1e-5f
#define EPS_AGG 1e-6f

typedef __attribute__((ext_vector_type(2))) float v2f;
typedef __attribute__((ext_vector_type(8))) float v8f;

// D = A(16x4) * B(4x16) + C, fp32, wave32. NEG/OPSEL modifiers off.
__device__ __forceinline__ v8f wmma_f32(v2f a, v2f b, v8f c) {
  return __builtin_amdgcn_wmma_f32_16x16x4_f32(
      /*neg_a=*/false, a, /*neg_b=*/false, b,
      /*c_mod=*/(short)0, c, /*reuse_a=*/false, /*reuse_b=*/false);
}

// ---------------------------------------------------------------------------
// Kernel 1: node GEMMs. One wave computes a 16x64 tile of one of {Ah,Bh,Dh,Eh}.
// Block = 8 waves = one 16-row tile across all 4 matrices x 2 column halves,
// so the A-tile (rows of h) is reused out of L0 across the block.
// grid = 50000/16 = 3125 blocks.
// ---------------------------------------------------------------------------
__global__ void node_gemm(const float* __restrict__ h,
                          const float* __restrict__ W0, const float* __restrict__ b0,
                          const float* __restrict__ W1, const float* __restrict__ b1,
                          const float* __restrict__ W2, const float* __restrict__ b2,
                          const float* __restrict__ W3, const float* __restrict__ b3,
                          float* __restrict__ O0, float* __restrict__ O1,
                          float* __restrict__ O2, float* __restrict__ O3) {
  const int wid  = threadIdx.x >> 5;
  const int lane = threadIdx.x & 31;
  const int gw   = blockIdx.x * 8 + wid;       // 0..24999
  const int row_tile = gw >> 3;
  const int sub  = gw & 7;
  const int mat  = sub >> 1;
  const int half = sub & 1;

  const float* W    = (mat == 0) ? W0 : (mat == 1) ? W1 : (mat == 2) ? W2 : W3;
  const float* bias = (mat == 0) ? b0 : (mat == 1) ? b1 : (mat == 2) ? b2 : b3;
  float*       O    = (mat == 0) ? O0 : (mat == 1) ? O1 : (mat == 2) ? O2 : O3;

  const int m0   = row_tile * 16;
  const int n0   = half * 64;
  const int lm   = lane & 15;
  const int hi   = lane >> 4;
  const int koff = hi * 2;

  v8f acc[4];
#pragma unroll
  for (int t = 0; t < 4; ++t) {
    const float bv = bias[n0 + t * 16 + lm];
#pragma unroll
    for (int v = 0; v < 8; ++v) acc[t][v] = bv;
  }

  const float* arow = h + (size_t)(m0 + lm) * DD;
#pragma unroll 4
  for (int k = 0; k < DD; k += 4) {
    const v2f a = *(const v2f*)(arow + k + koff);
#pragma unroll
    for (int t = 0; t < 4; ++t) {
      const int n = n0 + t * 16 + lm;
      v2f b;
      b[0] = W[(size_t)(k + koff) * DD + n];
      b[1] = W[(size_t)(k + koff + 1) * DD + n];
      acc[t] = wmma_f32(a, b, acc[t]);
    }
  }

#pragma unroll
  for (int t = 0; t < 4; ++t) {
    const int n = n0 + t * 16 + lm;
#pragma unroll
    for (int v = 0; v < 8; ++v) {
      const int m = m0 + v + 8 * hi;
      O[(size_t)m * DD + n] = acc[t][v];
    }
  }
}

// ---------------------------------------------------------------------------
// Kernel 2: edge GEMM (Ce) fused with edge update.
// One wave computes a 16-edge x 64-feature tile of Ce via WMMA, then:
//   x = Ce + Dh[src] + Eh[dst]   -> e_new (stored for BN pass)
//   sigma = sigmoid(x)           -> atomic scatter into sum_sigma{,_h}[dst]
//   BN stats of e_new reduced through LDS, one global atomic per feature/block
// grid covers 500000/16 = 31250 tiles * 2 halves = 62500 waves.
// ---------------------------------------------------------------------------
__global__ void edge_kernel(const float* __restrict__ e,
                            const int* __restrict__ src, const int* __restrict__ dst,
                            const float* __restrict__ Cw, const float* __restrict__ Cb,
                            const float* __restrict__ Bh, const float* __restrict__ Dh,
                            const float* __restrict__ Eh,
                            float* __restrict__ e_new,
                            float* __restrict__ acc_sh, float* __restrict__ acc_s,
                            float* __restrict__ e_sum, float* __restrict__ e_sumsq) {
  __shared__ float s_sum[DD];
  __shared__ float s_sq[DD];
  for (int i = threadIdx.x; i < DD; i += blockDim.x) { s_sum[i] = 0.f; s_sq[i] = 0.f; }
  __syncthreads();

  const int wid  = threadIdx.x >> 5;
  const int lane = threadIdx.x & 31;
  const int gw   = blockIdx.x * 8 + wid;
  const int tile = gw >> 1;          // wave-uniform -> EXEC stays all-ones
  const int half = gw & 1;

  if (tile < NEDGES / 16) {
    const int e0   = tile * 16;
    const int n0   = half * 64;
    const int lm   = lane & 15;
    const int hi   = lane >> 4;
    const int koff = hi * 2;

    v8f acc[4];
#pragma unroll
    for (int t = 0; t < 4; ++t) {
      const float bv = Cb[n0 + t * 16 + lm];
#pragma unroll
      for (int v = 0; v < 8; ++v) acc[t][v] = bv;
    }

    const float* arow = e + (size_t)(e0 + lm) * DD;
#pragma unroll 4
    for (int k = 0; k < DD; k += 4) {
      const v2f a = *(const v2f*)(arow + k + koff);
#pragma unroll
      for (int t = 0; t < 4; ++t) {
        const int n = n0 + t * 16 + lm;
        v2f b;
        b[0] = Cw[(size_t)(k + koff) * DD + n];
        b[1] = Cw[(size_t)(k + koff + 1) * DD + n];
        acc[t] = wmma_f32(a, b, acc[t]);
      }
    }

#pragma unroll
    for (int t = 0; t < 4; ++t) {
      const int n = n0 + t * 16 + lm;
      float psum = 0.f, psq = 0.f;
#pragma unroll
      for (int v = 0; v < 8; ++v) {
        const int m   = v + 8 * hi;
        const int eid = e0 + m;
        const int s   = src[eid];
        const int d   = dst[eid];
        const float x = acc[t][v] + Dh[(size_t)s * DD + n] + Eh[(size_t)d * DD + n];
        e_new[(size_t)eid * DD + n] = x;
        psum += x;
        psq  += x * x;
        const float sg = 1.0f / (1.0f + __expf(-x));
        atomicAdd(&acc_s[(size_t)d * DD + n], sg);
        atomicAdd(&acc_sh[(size_t)d * DD + n], sg * Bh[(size_t)s * DD + n]);
      }
      atomicAdd(&s_sum[n], psum);
      atomicAdd(&s_sq[n], psq);
    }
  }
  __syncthreads();
  for (int i = threadIdx.x; i < DD; i += blockDim.x) {
    atomicAdd(&e_sum[i], s_sum[i]);
    atomicAdd(&e_sumsq[i], s_sq[i]);
  }
}

// ---------------------------------------------------------------------------
// Kernel 3: h_new = Ah + sum_sigma_h / (sum_sigma + eps), plus BN stats of h.
// One thread per element; grid = 50000*128/256 = 25000 blocks.
// ---------------------------------------------------------------------------
__global__ void node_combine(const float* __restrict__ Ah,
                             const float* __restrict__ acc_sh,
                             const float* __restrict__ acc_s,
                             float* __restrict__ h_new,
                             float* __restrict__ h_sum, float* __restrict__ h_sumsq) {
  __shared__ float s_sum[DD];
  __shared__ float s_sq[DD];
  for (int i = threadIdx.x; i < DD; i += blockDim.x) { s_sum[i] = 0.f; s_sq[i] = 0.f; }
  __syncthreads();

  const size_t idx = (size_t)blockIdx.x * blockDim.x + threadIdx.x;
  const float x = Ah[idx] + acc_sh[idx] / (acc_s[idx] + EPS_AGG);
  h_new[idx] = x;
  const int n = threadIdx.x & (DD - 1);   // blockDim multiple of 128
  atomicAdd(&s_sum[n], x);
  atomicAdd(&s_sq[n], x * x);

  __syncthreads();
  if (threadIdx.x < DD) {
    atomicAdd(&h_sum[threadIdx.x], s_sum[threadIdx.x]);
    atomicAdd(&h_sumsq[threadIdx.x], s_sq[threadIdx.x]);
  }
}

// ---------------------------------------------------------------------------
// Kernel 4: out = x_in + relu(BN(x_new)) using biased batch stats.
// ---------------------------------------------------------------------------
__global__ void bn_relu_residual(const float* __restrict__ xin,
                                 const float* __restrict__ xnew,
                                 const float* __restrict__ sum,
                                 const float* __restrict__ sumsq,
                                 const float* __restrict__ gamma,
                                 const float* __restrict__ beta,
                                 float invN, float* __restrict__ out, size_t total) {
  const size_t idx = (size_t)blockIdx.x * blockDim.x + threadIdx.x;
  if (idx >= total) return;
  const int n = (int)(idx & (DD - 1));
  const float mean = sum[n] * invN;
  const float var  = sumsq[n] * invN - mean * mean;
  const float x    = xnew[idx];
  float y = (x - mean) * rsqrtf(var + EPS_BN) * gamma[n] + beta[n];
  y = fmaxf(y, 0.f);
  out[idx] = xin[idx] + y;
}

// ---------------------------------------------------------------------------
extern "C" void kernel_launch(void* const* d_in, const int* in_sizes, int n_in,
                              void* d_out, int out_size, void* d_ws, size_t ws_size,
                              hipStream_t stream) {
  const float* h       = (const float*)d_in[0];
  const float* e       = (const float*)d_in[1];
  const int*   src     = (const int*)d_in[2];
  const int*   dst     = (const int*)d_in[3];
  const float* Aw      = (const float*)d_in[4];
  const float* Ab      = (const float*)d_in[5];
  const float* Bw      = (const float*)d_in[6];
  const float* Bb      = (const float*)d_in[7];
  const float* Cw      = (const float*)d_in[8];
  const float* Cb      = (const float*)d_in[9];
  const float* Dw      = (const float*)d_in[10];
  const float* Db      = (const float*)d_in[11];
  const float* Ew      = (const float*)d_in[12];
  const float* Eb      = (const float*)d_in[13];
  const float* gamma_h = (const float*)d_in[14];
  const float* beta_h  = (const float*)d_in[15];
  const float* gamma_e = (const float*)d_in[16];
  const float* beta_e  = (const float*)d_in[17];

  const size_t NH = (size_t)NNODES * DD;   // 6.4M
  const size_t NE = (size_t)NEDGES * DD;   // 64M

  float* Ah     = (float*)d_ws;
  float* Bh     = Ah + NH;
  float* Dh     = Bh + NH;
  float* Eh     = Dh + NH;
  float* acc_sh = Eh + NH;
  float* acc_s  = acc_sh + NH;
  float* h_new  = acc_s + NH;
  float* h_sum  = h_new + NH;       // 128
  float* h_sq   = h_sum + DD;       // 128
  float* e_sum  = h_sq + DD;        // 128
  float* e_sq   = e_sum + DD;       // 128
  float* e_new  = e_sq + DD;        // 64M

  float* h_out = (float*)d_out;
  float* e_out = h_out + NH;

  // Zero accumulators + stats (graph-capture-safe async memsets).
  hipMemsetAsync(acc_sh, 0, 2 * NH * sizeof(float), stream);          // acc_sh, acc_s
  hipMemsetAsync(h_sum, 0, 4 * DD * sizeof(float), stream);           // all stats

  // 1) Ah, Bh, Dh, Eh
  node_gemm<<<NNODES / 16, 256, 0, stream>>>(h, Aw, Ab, Bw, Bb, Dw, Db, Ew, Eb,
                                             Ah, Bh, Dh, Eh);
  // 2) e_new, sigma scatters, e BN stats
  const int edge_waves  = (NEDGES / 16) * 2;                  // 62500
  const int edge_blocks = (edge_waves + 7) / 8;               // 7813
  edge_kernel<<<edge_blocks, 256, 0, stream>>>(e, src, dst, Cw, Cb, Bh, Dh, Eh,
                                               e_new, acc_sh, acc_s, e_sum, e_sq);
  // 3) h_new + h BN stats
  node_combine<<<(int)(NH / 256), 256, 0, stream>>>(Ah, acc_sh, acc_s, h_new,
                                                    h_sum, h_sq);
  // 4) finalize h and e
  bn_relu_residual<<<(int)(NH / 256), 256, 0, stream>>>(
      h, h_new, h_sum, h_sq, gamma_h, beta_h, 1.0f / (float)NNODES, h_out, NH);
  bn_relu_residual<<<(int)(NE / 256), 256, 0, stream>>>(
      e, e_new, e_sum, e_sq, gamma_e, beta_e, 1.0f / (float)NEDGES, e_out, NE);
}